// ExampleModel_19490561590024
// MI455X (gfx1250) — compile-verified
//
#include <hip/hip_runtime.h>

// ---------------- problem constants (match reference) ----------------
#define HOP    128
#define WIN    512
#define NBIN   257              // WIN/2 + 1
#define BATCH  8
#define TLEN   262144
#define NFRM   2045             // 1 + (TLEN - WIN)/HOP
#define BF     (BATCH * NFRM)   // 16360 total frames (GEMM M dimension)
#define MTILES 1024             // padded so each 8-wave block is full
#define MPAD   (MTILES * 16)    // 16384
#define KPAD   260              // NBIN padded to multiple of 4 (iSTFT K dim)
#define NPADF  272              // 17 tiles * 16 (STFT N dim padded)

// ---------------- workspace layout (float offsets) ----------------
// everything is L2-resident (~70 MB < 192 MB global L2)
#define OFF_TCF 0u                                   // [WIN][NPADF] cos fwd
#define OFF_TSF (OFF_TCF + WIN * NPADF)              // [WIN][NPADF] -sin fwd
#define OFF_TCI (OFF_TSF + WIN * NPADF)              // [KPAD][WIN]  irfft cos basis
#define OFF_TSI (OFF_TCI + KPAD * WIN)               // [KPAD][WIN]  irfft -sin basis
#define OFF_RE  (OFF_TSI + KPAD * WIN)               // [MPAD][KPAD]
#define OFF_IM  (OFF_RE + (size_t)MPAD * KPAD)       // [MPAD][KPAD]
#define OFF_FR  (OFF_IM + (size_t)MPAD * KPAD)       // [BF][WIN] windowed synth frames

typedef float        v2f __attribute__((ext_vector_type(2)));
typedef float        v8f __attribute__((ext_vector_type(8)));
typedef unsigned int v4u __attribute__((ext_vector_type(4)));
typedef int          v4i __attribute__((ext_vector_type(4)));
typedef int          v8i __attribute__((ext_vector_type(8)));

// ---------------------------------------------------------------
// TDM: async 2D tile load Global -> LDS (Tensor Data Mover, ISA ch.8).
// D# group0: count=1, lds byte addr, 57-bit global addr, type=2 ("image").
// D# group1: data_size=4B (code 2), tensor_dim0/1, tile_dim0/1, dim0 stride.
// Groups 2/3 zero (2-D tensor), trailing group zero, cpol 0.
// ---------------------------------------------------------------
__device__ __forceinline__ void tdm_load_2d(unsigned lds_off, const float* gptr,
                                            unsigned dim0, unsigned dim1,
                                            unsigned tile0, unsigned tile1,
                                            unsigned stride0) {
    const unsigned long long ga = (unsigned long long)(uintptr_t)gptr;
    v4u g0;
    g0.x = 1u;                                            // count=1, user mode
    g0.y = lds_off;                                       // LDS byte address
    g0.z = (unsigned)(ga & 0xffffffffull);                // global addr [31:0]
    g0.w = (unsigned)((ga >> 32) & 0x01ffffffull)         // global addr [56:32]
         | (2u << 30);                                    // type = 2
    v8i g1;
    g1[0] = (int)(2u << 16);                              // data_size = 4 bytes
    g1[1] = (int)((dim0 & 0xffffu) << 16);                // tensor_dim0 [15:0]
    g1[2] = (int)((dim0 >> 16) | ((dim1 & 0xffffu) << 16));
    g1[3] = (int)((dim1 >> 16) | (tile0 << 16));          // tile_dim0
    g1[4] = (int)(tile1 & 0xffffu);                       // tile_dim1 (tile_dim2=0)
    g1[5] = (int)stride0;                                 // tensor_dim0_stride [31:0]
    g1[6] = 0;
    g1[7] = 0;
    const v4i z4 = {};
    const v8i z8 = {};
    __builtin_amdgcn_tensor_load_to_lds(g0, g1, z4, z4, z8, 0);
}

// ---------------------------------------------------------------
// Kernel 1: DFT / inverse-DFT basis tables.
// (w*k) mod 512 is exact in integers; hardware V_SIN/COS_F32 take the
// argument in revolutions: __builtin_amdgcn_cosf(x) == cos(2*pi*x).
// Columns k >= NBIN are zeroed so padded GEMM lanes contribute nothing.
// ---------------------------------------------------------------
__global__ void build_tables(float* __restrict__ ws) {
    const int idx = blockIdx.x * blockDim.x + threadIdx.x;
    const float inv = 1.0f / (float)WIN;

    if (idx < WIN * NPADF) {
        const int w = idx / NPADF;
        const int k = idx % NPADF;
        const int r = (w * k) & (WIN - 1);
        const float rev = (float)r * inv;
        const float c =  __builtin_amdgcn_cosf(rev);
        const float s = -__builtin_amdgcn_sinf(rev);      // rfft imag sign
        ws[OFF_TCF + idx] = (k < NBIN) ? c : 0.0f;
        ws[OFF_TSF + idx] = (k < NBIN) ? s : 0.0f;
    }
    if (idx < KPAD * WIN) {
        const int k = idx / WIN;
        const int w = idx % WIN;
        const int r = (w * k) & (WIN - 1);
        const float rev  = (float)r * inv;
        const float coef = ((k == 0 || k == NBIN - 1) ? 1.0f : 2.0f) * inv;
        const float ci =  coef * __builtin_amdgcn_cosf(rev);
        const float si = -coef * __builtin_amdgcn_sinf(rev);
        ws[OFF_TCI + idx] = (k < NBIN) ? ci : 0.0f;
        ws[OFF_TSI + idx] = (k < NBIN) ? si : 0.0f;
    }
}

// ---------------------------------------------------------------
// Kernel 2: STFT as WMMA f32 16x16x4 GEMM, 8 waves/block.
// All 8 waves share one N-tile; the DFT basis panel (512x16 cos + sin,
// 64 KB) is TDM-DMA'd into LDS once per block, then read via ds_load.
// A-operand (audio*hann) loads are unconditional (clamped addr * mask).
// ---------------------------------------------------------------
__global__ void __launch_bounds__(256)
stft_wmma(const float* __restrict__ audio,
          const float* __restrict__ window,
          float* __restrict__ ws) {
    __shared__ float panel[2 * WIN * 16];          // [cos | sin] : 64 KB
    float* panC = panel;
    float* panS = panel + WIN * 16;

    const float* tcf = ws + OFF_TCF;
    const float* tsf = ws + OFF_TSF;
    float* RE = ws + OFF_RE;
    float* IM = ws + OFF_IM;

    const int tid   = threadIdx.x;
    const int lane  = tid & 31;
    const int wave  = tid >> 5;                    // 0..7
    const int tileN = blockIdx.x;                  // 0..16  (k bins)
    const int tileM = blockIdx.y * 8 + wave;       // 0..1023

    // one TDM per panel, issued by wave 0 only (EXEC-independent DMA)
    if (wave == 0) {
        tdm_load_2d((unsigned)(uintptr_t)panC, tcf + tileN * 16,
                    NPADF, WIN, 16, WIN, NPADF);
        tdm_load_2d((unsigned)(uintptr_t)panS, tsf + tileN * 16,
                    NPADF, WIN, 16, WIN, NPADF);
        __builtin_amdgcn_s_wait_tensorcnt(0);
    }
    __syncthreads();

    const int mrow  = lane & 15;
    const int khalf = (lane >> 4) << 1;            // 0 or 2
    const int ncol  = lane & 15;

    const int   m0     = tileM * 16 + mrow;
    const float msk    = (m0 < BF) ? 1.0f : 0.0f;  // pad rows -> zero, no branch
    const int   mc     = (m0 < BF) ? m0 : 0;       // clamped (always-legal addr)
    const int   b      = mc / NFRM;
    const int   f      = mc - b * NFRM;
    const long  abase  = (long)b * TLEN + (long)f * HOP;

    v8f cre = {};
    v8f cim = {};

    for (int kk = 0; kk < WIN; kk += 4) {
        const int w0 = kk + khalf;
        v2f a, bc, bs;
        a.x = audio[abase + w0]     * window[w0]     * msk;
        a.y = audio[abase + w0 + 1] * window[w0 + 1] * msk;
        bc.x = panC[(w0)     * 16 + ncol];
        bc.y = panC[(w0 + 1) * 16 + ncol];
        bs.x = panS[(w0)     * 16 + ncol];
        bs.y = panS[(w0 + 1) * 16 + ncol];
        cre = __builtin_amdgcn_wmma_f32_16x16x4_f32(false, a, false, bc, (short)0, cre, false, false);
        cim = __builtin_amdgcn_wmma_f32_16x16x4_f32(false, a, false, bs, (short)0, cim, false, false);
    }

    const int mhalf = (lane >> 4) << 3;            // 0 or 8
    const int n     = tileN * 16 + ncol;
    if (n < KPAD) {
        for (int r = 0; r < 8; ++r) {
            const int m = tileM * 16 + mhalf + r;  // < MPAD always (pad rows store 0)
            RE[(long)m * KPAD + n] = cre[r];
            IM[(long)m * KPAD + n] = cim[r];
        }
    }
}

// ---------------------------------------------------------------
// Kernel 3: iSTFT as WMMA f32 16x16x4 GEMM, 8 waves/block.
//   y[m,w] = sum_k RE[m,k]*ci[k,w] + IM[m,k]*si[k,w]   (exact irfft fold)
// Basis panels (260x16 each) TDM-staged into LDS; synthesis window
// fused at store.
// ---------------------------------------------------------------
__global__ void __launch_bounds__(256)
istft_wmma(const float* __restrict__ window,
           float* __restrict__ ws) {
    __shared__ float panel[2 * KPAD * 16];         // [cos | sin] : 33.3 KB
    float* panC = panel;
    float* panS = panel + KPAD * 16;

    const float* RE  = ws + OFF_RE;
    const float* IM  = ws + OFF_IM;
    const float* tci = ws + OFF_TCI;
    const float* tsi = ws + OFF_TSI;
    float* FR = ws + OFF_FR;

    const int tid   = threadIdx.x;
    const int lane  = tid & 31;
    const int wave  = tid >> 5;
    const int tileN = blockIdx.x;                  // 0..31 (time cols)
    const int tileM = blockIdx.y * 8 + wave;       // 0..1023

    if (wave == 0) {
        tdm_load_2d((unsigned)(uintptr_t)panC, tci + tileN * 16,
                    WIN, KPAD, 16, KPAD, WIN);
        tdm_load_2d((unsigned)(uintptr_t)panS, tsi + tileN * 16,
                    WIN, KPAD, 16, KPAD, WIN);
        __builtin_amdgcn_s_wait_tensorcnt(0);
    }
    __syncthreads();

    const int mrow  = lane & 15;
    const int khalf = (lane >> 4) << 1;
    const int ncol  = lane & 15;

    const int m0 = tileM * 16 + mrow;              // < MPAD (RE/IM padded)
    const int n0 = tileN * 16 + ncol;

    v8f acc = {};
    for (int kk = 0; kk < KPAD; kk += 4) {
        const int k0 = kk + khalf;
        v2f are, aim, bc, bs;
        are.x = RE[(long)m0 * KPAD + k0];
        are.y = RE[(long)m0 * KPAD + k0 + 1];
        aim.x = IM[(long)m0 * KPAD + k0];
        aim.y = IM[(long)m0 * KPAD + k0 + 1];
        bc.x = panC[(k0)     * 16 + ncol];
        bc.y = panC[(k0 + 1) * 16 + ncol];
        bs.x = panS[(k0)     * 16 + ncol];
        bs.y = panS[(k0 + 1) * 16 + ncol];
        acc = __builtin_amdgcn_wmma_f32_16x16x4_f32(false, are, false, bc, (short)0, acc, false, false);
        acc = __builtin_amdgcn_wmma_f32_16x16x4_f32(false, aim, false, bs, (short)0, acc, false, false);
    }

    const int   mhalf = (lane >> 4) << 3;
    const float wcol  = window[n0];
    for (int r = 0; r < 8; ++r) {
        const int m = tileM * 16 + mhalf + r;
        if (m < BF) {
            FR[(long)m * WIN + n0] = acc[r] * wcol;
        }
    }
}

// ---------------------------------------------------------------
// Kernel 4: overlap-add + wsq normalization (pure bandwidth, HBM floor
// ~16 MB of traffic at 23.3 TB/s). <= 4 frames cover each sample
// (hop = win/4); wsq recomputed inline, reproducing max(wsq,1e-8) edges.
// ---------------------------------------------------------------
__global__ void overlap_add(const float* __restrict__ window,
                            const float* __restrict__ frames,   // ws + OFF_FR
                            float* __restrict__ out) {
    const long idx = (long)blockIdx.x * blockDim.x + threadIdx.x;
    if (idx >= (long)BATCH * TLEN) return;
    const int b = (int)(idx / TLEN);
    const int n = (int)(idx % TLEN);

    int fmax = n / HOP;               if (fmax > NFRM - 1) fmax = NFRM - 1;
    int fmin = (n - WIN + HOP) / HOP; if (fmin < 0)        fmin = 0;

    float acc = 0.0f, wsq = 0.0f;
    for (int f = fmin; f <= fmax; ++f) {
        const int   w  = n - f * HOP;
        const float wv = window[w];
        acc += frames[((long)b * NFRM + f) * WIN + w];
        wsq += wv * wv;
    }
    out[idx] = acc / fmaxf(wsq, 1e-8f);
}

// ---------------------------------------------------------------
// Launch: build tables -> STFT GEMM -> iSTFT GEMM -> overlap-add.
// The magnitude + LSTM of the reference are exact dead code
// (mask_re = x*0+1, mask_im = x*0) and are skipped entirely.
// ---------------------------------------------------------------
extern "C" void kernel_launch(void* const* d_in, const int* in_sizes, int n_in,
                              void* d_out, int out_size, void* d_ws, size_t ws_size,
                              hipStream_t stream) {
    (void)in_sizes; (void)n_in; (void)out_size; (void)ws_size;
    const float* audio  = (const float*)d_in[0];
    const float* window = (const float*)d_in[1];
    float* ws  = (float*)d_ws;
    float* out = (float*)d_out;

    {
        const int n = WIN * NPADF;   // covers both table regions
        build_tables<<<(n + 255) / 256, 256, 0, stream>>>(ws);
    }
    {
        dim3 grid(17, MTILES / 8);   // N-tiles x M-groups, 8 waves/block
        stft_wmma<<<grid, 256, 0, stream>>>(audio, window, ws);
    }
    {
        dim3 grid(WIN / 16, MTILES / 8);
        istft_wmma<<<grid, 256, 0, stream>>>(window, ws);
    }
    {
        const long total = (long)BATCH * TLEN;
        overlap_add<<<(int)((total + 255) / 256), 256, 0, stream>>>(
            window, ws + OFF_FR, out);
    }
}